// Sonata3DSegLevelEncoder_57569741635786
// MI455X (gfx1250) — compile-verified
//
#include <hip/hip_runtime.h>
#include <hip/hip_bf16.h>

typedef __attribute__((ext_vector_type(2))) float v2f;
typedef __attribute__((ext_vector_type(8))) float v8f;

#define NPTS 60000
#define NSEG 1600      // B * SEG = 4 * 400
#define HID  768

// ---------------------------------------------------------------- zero scratch
__global__ void zero_kernel(float4* __restrict__ p, long long n4) {
    long long i = (long long)blockIdx.x * blockDim.x + threadIdx.x;
    if (i < n4) p[i] = make_float4(0.f, 0.f, 0.f, 0.f);
}

// ---------------------------------------------------------------- segment counts
__global__ void count_kernel(const int* __restrict__ batch,
                             const int* __restrict__ p2s,
                             const int* __restrict__ msp,
                             float* __restrict__ cnt) {
    int n = blockIdx.x * blockDim.x + threadIdx.x;
    if (n < NPTS) {
        int seg = batch[n] * msp[0] + p2s[n];
        atomicAdd(cnt + seg, 1.0f);
    }
}

// ---------------------------------------------------------------- masked gather + segment scatter-sum
// One block per point: coalesced float4 gather of the feats row, atomic add into
// the segment accumulator (31MB total -> L2 resident on MI455X's 192MB L2).
template <int C>
__global__ void scatter_sum_kernel(const float* __restrict__ feats,
                                   const int* __restrict__ idx,
                                   const int* __restrict__ mask,
                                   const int* __restrict__ batch,
                                   const int* __restrict__ p2s,
                                   const int* __restrict__ msp,
                                   float* __restrict__ ssum) {
    const int n = blockIdx.x;
    if (!mask[n]) return;
    const int seg = batch[n] * msp[0] + p2s[n];
    const float4* __restrict__ src = (const float4*)(feats + (size_t)idx[n] * C);
    float* __restrict__ dst = ssum + (size_t)seg * C;
    for (int c4 = threadIdx.x; c4 < C / 4; c4 += blockDim.x) {
        float4 f = src[c4];
        atomicAdd(dst + c4 * 4 + 0, f.x);
        atomicAdd(dst + c4 * 4 + 1, f.y);
        atomicAdd(dst + c4 * 4 + 2, f.z);
        atomicAdd(dst + c4 * 4 + 3, f.w);
    }
}

// ---------------------------------------------------------------- WMMA GEMM: (ssum/cnt)[1600,C] x W[C,768] + b
// One wave32 per 16x16 output tile; K stepped by 4 via v_wmma_f32_16x16x4_f32.
// A layout (16x4 f32, ISA 7.12.2): lane l, VGPR r holds A[M = l%16][K = r + 2*(l>=16)].
// C/D layout: lane l, VGPR r holds D[M = r + 8*(l>=16)][N = l%16].
template <int C>
__global__ void gemm_kernel(const float* __restrict__ ssum,
                            const float* __restrict__ cnt,
                            const float* __restrict__ W,
                            const float* __restrict__ bias,
                            float* __restrict__ out) {
    const int lane = threadIdx.x & 31;
    const int wid  = threadIdx.x >> 5;
    const int tile = blockIdx.x * 8 + wid;   // 0..4799 exactly (600 blocks x 8 waves)
    const int tm   = tile / 48;              // 0..99  (1600/16 row tiles)
    const int tn   = tile % 48;              // 0..47  (768/16 col tiles)
    const int half = lane >> 4;
    const int l16  = lane & 15;
    const int col  = tn * 16 + l16;

    // fused scatter_mean division: scale A rows by 1/max(cnt,1)
    const float rc = 1.0f / fmaxf(cnt[tm * 16 + l16], 1.0f);
    const float* __restrict__ srow = ssum + (size_t)(tm * 16 + l16) * C;

    v8f acc = {};
#pragma unroll 4
    for (int k0 = 0; k0 < C; k0 += 4) {
        const int kk = k0 + 2 * half;
        float2 av = *(const float2*)(srow + kk);
        v2f a;
        a.x = av.x * rc;
        a.y = av.y * rc;
        v2f b;
        b.x = W[(size_t)kk * HID + col];
        b.y = W[(size_t)(kk + 1) * HID + col];
        acc = __builtin_amdgcn_wmma_f32_16x16x4_f32(
            /*neg_a=*/false, a, /*neg_b=*/false, b,
            /*c_mod=*/(short)0, acc, /*reuse_a=*/false, /*reuse_b=*/false);
    }

    const float bv = bias[col];
    float* __restrict__ orow = out + (size_t)(tm * 16 + 8 * half) * HID + col;
#pragma unroll
    for (int r = 0; r < 8; ++r) orow[(size_t)r * HID] = acc[r] + bv;
}

// ---------------------------------------------------------------- in-place LayerNorm over H=768, block per row
__global__ void ln_kernel(float* __restrict__ y,
                          const float* __restrict__ g,
                          const float* __restrict__ be) {
    __shared__ float shs[8], shq[8], stat[2];
    float* __restrict__ p = y + (size_t)blockIdx.x * HID;
    const int t = threadIdx.x;

    float v0 = p[t], v1 = p[t + 256], v2 = p[t + 512];
    float s = v0 + v1 + v2;
    float q = v0 * v0 + v1 * v1 + v2 * v2;
#pragma unroll
    for (int o = 16; o > 0; o >>= 1) {
        s += __shfl_down(s, o, 32);
        q += __shfl_down(q, o, 32);
    }
    if ((t & 31) == 0) { shs[t >> 5] = s; shq[t >> 5] = q; }
    __syncthreads();
    if (t == 0) {
        float S = 0.f, Q = 0.f;
#pragma unroll
        for (int w = 0; w < 8; ++w) { S += shs[w]; Q += shq[w]; }
        float mu  = S * (1.0f / HID);
        float var = Q * (1.0f / HID) - mu * mu;
        stat[0] = mu;
        stat[1] = rsqrtf(var + 1e-5f);
    }
    __syncthreads();
    const float mu = stat[0], rs = stat[1];
    p[t]       = (v0 - mu) * rs * g[t]       + be[t];
    p[t + 256] = (v1 - mu) * rs * g[t + 256] + be[t + 256];
    p[t + 512] = (v2 - mu) * rs * g[t + 512] + be[t + 512];
}

// ---------------------------------------------------------------- launcher
extern "C" void kernel_launch(void* const* d_in, const int* in_sizes, int n_in,
                              void* d_out, int out_size, void* d_ws, size_t ws_size,
                              hipStream_t stream) {
    (void)in_sizes; (void)n_in; (void)out_size; (void)ws_size;

    const int* batch = (const int*)d_in[0];
    const int* p2s   = (const int*)d_in[1];
    const int* msp   = (const int*)d_in[2];   // max_seg scalar on device
    float* out = (float*)d_out;
    float* ws  = (float*)d_ws;

    const int Cs[5] = {512, 896, 1088, 1184, 1232};

    // workspace layout: cnt[1600] then per-scale segment sums [1600, C_i]
    float* cnt = ws;
    size_t off = NSEG;
    float* sums[5];
    for (int i = 0; i < 5; ++i) { sums[i] = ws + off; off += (size_t)NSEG * Cs[i]; }

    // outputs reversed: scale i lives at slot (4-i)
    float* outs[5];
    for (int i = 0; i < 5; ++i) outs[i] = out + (size_t)(4 - i) * NSEG * HID;

    // 1) zero scratch (off is a multiple of 4 floats: all C_i % 16 == 0)
    long long n4 = (long long)off / 4;
    zero_kernel<<<(unsigned)((n4 + 255) / 256), 256, 0, stream>>>((float4*)ws, n4);

    // 2) segment counts
    count_kernel<<<(NPTS + 255) / 256, 256, 0, stream>>>(batch, p2s, msp, cnt);

#define FE(i) ((const float*)d_in[3 + 7 * (i) + 0])
#define IX(i) ((const int*)  d_in[3 + 7 * (i) + 1])
#define MK(i) ((const int*)  d_in[3 + 7 * (i) + 2])
#define WW(i) ((const float*)d_in[3 + 7 * (i) + 3])
#define BV(i) ((const float*)d_in[3 + 7 * (i) + 4])
#define GG(i) ((const float*)d_in[3 + 7 * (i) + 5])
#define BE(i) ((const float*)d_in[3 + 7 * (i) + 6])

    // 3) masked gather + segment scatter-sum (the bandwidth-dominant stage)
    scatter_sum_kernel< 512><<<NPTS, 128, 0, stream>>>(FE(0), IX(0), MK(0), batch, p2s, msp, sums[0]);
    scatter_sum_kernel< 896><<<NPTS, 128, 0, stream>>>(FE(1), IX(1), MK(1), batch, p2s, msp, sums[1]);
    scatter_sum_kernel<1088><<<NPTS, 128, 0, stream>>>(FE(2), IX(2), MK(2), batch, p2s, msp, sums[2]);
    scatter_sum_kernel<1184><<<NPTS, 128, 0, stream>>>(FE(3), IX(3), MK(3), batch, p2s, msp, sums[3]);
    scatter_sum_kernel<1232><<<NPTS, 128, 0, stream>>>(FE(4), IX(4), MK(4), batch, p2s, msp, sums[4]);

    // 4) WMMA GEMM with fused mean + bias -> reversed output slots
    gemm_kernel< 512><<<600, 256, 0, stream>>>(sums[0], cnt, WW(0), BV(0), outs[0]);
    gemm_kernel< 896><<<600, 256, 0, stream>>>(sums[1], cnt, WW(1), BV(1), outs[1]);
    gemm_kernel<1088><<<600, 256, 0, stream>>>(sums[2], cnt, WW(2), BV(2), outs[2]);
    gemm_kernel<1184><<<600, 256, 0, stream>>>(sums[3], cnt, WW(3), BV(3), outs[3]);
    gemm_kernel<1232><<<600, 256, 0, stream>>>(sums[4], cnt, WW(4), BV(4), outs[4]);

    // 5) LayerNorm in place
    for (int i = 0; i < 5; ++i)
        ln_kernel<<<NSEG, 256, 0, stream>>>(outs[i], GG(i), BE(i));
}